// Purity_54829552501242
// MI455X (gfx1250) — compile-verified
//
#include <hip/hip_runtime.h>

typedef __attribute__((ext_vector_type(2))) float v2f;
typedef __attribute__((ext_vector_type(8))) float v8f;

#define ELEMS 4096   // 64*64 per matrix
#define TPB   256    // 8 wave32 per block

__global__ __launch_bounds__(TPB) void purity_frob_kernel(
    const float* __restrict__ re,
    const float* __restrict__ im,
    float* __restrict__ out)
{
    __shared__ float partial[TPB];

    const int b = blockIdx.x;
    const int t = threadIdx.x;

    const float4* rea = reinterpret_cast<const float4*>(re + (size_t)b * ELEMS);
    const float4* ima = reinterpret_cast<const float4*>(im + (size_t)b * ELEMS);

    // 1024 float4 per matrix, 256 threads -> 4 float4 each from re and im.
    // Consecutive lanes touch consecutive 16B chunks: coalesced global_load_b128.
    float acc = 0.0f;
#pragma unroll
    for (int i = 0; i < 4; ++i) {
        float4 r = rea[t + i * TPB];
        float4 m = ima[t + i * TPB];
        acc = fmaf(r.x, r.x, acc);
        acc = fmaf(r.y, r.y, acc);
        acc = fmaf(r.z, r.z, acc);
        acc = fmaf(r.w, r.w, acc);
        acc = fmaf(m.x, m.x, acc);
        acc = fmaf(m.y, m.y, acc);
        acc = fmaf(m.z, m.z, acc);
        acc = fmaf(m.w, m.w, acc);
    }

    partial[t] = acc;
    __syncthreads();

    // Wave 0 (threads 0..31, wave-uniform guard -> EXEC all-ones inside)
    if (t < 32) {
        // Fold 256 partials -> 64 values, 2 per lane: the f32 16x4 A-matrix.
        float s0 = partial[t]       + partial[t + 32]  + partial[t + 64]  + partial[t + 96];
        float s1 = partial[t + 128] + partial[t + 160] + partial[t + 192] + partial[t + 224];

        v2f a;  a[0] = s0;  a[1] = s1;      // A: 16x4 f32, every partial appears exactly once
        v2f bo; bo[0] = 1.0f; bo[1] = 1.0f; // B: 4x16 all-ones -> D[i][j] = rowsum_i
        v8f c = {};

        // v_wmma_f32_16x16x4_f32: D = A x ones + 0
        v8f d = __builtin_amdgcn_wmma_f32_16x16x4_f32(
            /*neg_a=*/false, a, /*neg_b=*/false, bo,
            /*c_mod=*/(short)0, c, /*reuse_a=*/false, /*reuse_b=*/false);

        // Lane l<16 holds D rows 0..7 (col l), lane l>=16 holds rows 8..15 (col l-16).
        // Sum the 8 D registers, then combine halves with ds_swizzle SWAPX16.
        float rsum = d[0] + d[1] + d[2] + d[3] + d[4] + d[5] + d[6] + d[7];

        int other = __builtin_amdgcn_ds_swizzle(__float_as_int(rsum), 0x401F); // xor 0x10
        float total = rsum + __int_as_float(other);

        if (t == 0) out[b] = total;
    }
}

extern "C" void kernel_launch(void* const* d_in, const int* in_sizes, int n_in,
                              void* d_out, int out_size, void* d_ws, size_t ws_size,
                              hipStream_t stream) {
    const float* re = (const float*)d_in[0];
    const float* im = (const float*)d_in[1];
    float* out = (float*)d_out;

    const int batches = out_size; // 8192 density matrices
    purity_frob_kernel<<<dim3(batches), dim3(TPB), 0, stream>>>(re, im, out);
}